// EncDecAttention_1726576856250
// MI455X (gfx1250) — compile-verified
//
#include <hip/hip_runtime.h>
#include <math.h>

// ---------------------------------------------------------------------------
// Problem constants (from reference setup_inputs)
// ---------------------------------------------------------------------------
#define BB 2
#define NN 1024
#define MM 2048
#define CC 1024
#define HH 16
#define HD 64                     // head dim = C / H
#define LOGIT_MAX_C 4.6051701859880914f  // ln(1/0.01)

typedef float v2f __attribute__((ext_vector_type(2)));
typedef float v8f __attribute__((ext_vector_type(8)));

// Pointer types for the async global->LDS builtin (per hipcc diagnostic:
// param0 is "__attribute__((vector_size(16))) int __device__ *", i.e. v4i in AS1)
typedef int v4i __attribute__((vector_size(16)));
typedef __attribute__((address_space(1))) v4i* as1_v4i_ptr;
typedef __attribute__((address_space(3))) v4i* as3_v4i_ptr;

static __device__ __forceinline__ v8f wmma_f32(v2f a, v2f b, v8f c) {
  // D = A(16x4 f32) * B(4x16 f32) + C(16x16 f32)
  return __builtin_amdgcn_wmma_f32_16x16x4_f32(
      /*neg_a=*/false, a, /*neg_b=*/false, b,
      /*c_mod=*/(short)0, c, /*reuse_a=*/false, /*reuse_b=*/false);
}

// Async global -> LDS copy (16B per lane), tracked by ASYNCcnt.
static __device__ __forceinline__ void async_copy_b128(void* lds, const void* gaddr) {
#if __has_builtin(__builtin_amdgcn_global_load_async_to_lds_b128)
  __builtin_amdgcn_global_load_async_to_lds_b128(
      (as1_v4i_ptr)gaddr, (as3_v4i_ptr)lds, 0, 0);
#else
  const unsigned loff =
      (unsigned)(unsigned long long)(__attribute__((address_space(3))) void*)lds;
  asm volatile("global_load_async_to_lds_b128 %0, %1, off"
               :: "v"(loff), "v"(gaddr) : "memory");
#endif
}

static __device__ __forceinline__ void wait_async0() {
#if __has_builtin(__builtin_amdgcn_s_wait_asynccnt)
  __builtin_amdgcn_s_wait_asynccnt(0);
#else
  asm volatile("s_wait_asynccnt 0x0" ::: "memory");
#endif
}

// ---------------------------------------------------------------------------
// Wave-level GEMM: Y[r,c] = sum_k X[r,k] * W[c,k] (+ bias[c])
// One wave -> one 16x16 tile. 4 waves/block cover 64 consecutive columns.
// A layout (f32 16x4): lane lo holds row M=lo, hi selects K pair {2hi,2hi+1}.
// B layout mirrors A with lanes as N.
// C/D layout: vgpr r holds row (r + 8*hi), lane lo holds column lo.
// ---------------------------------------------------------------------------
__global__ __launch_bounds__(128) void gemm16_nt(
    const float* __restrict__ X, const float* __restrict__ W,
    const float* __restrict__ bias, float* __restrict__ Y,
    int K, int Cout)
{
  const int tid  = threadIdx.x;
  const int wave = tid >> 5;
  const int lane = tid & 31;
  const int lo   = lane & 15;
  const int hi   = lane >> 4;
  const int col0 = (blockIdx.x * 4 + wave) * 16;
  const int row0 = blockIdx.y * 16;

  const float* xr = X + (size_t)(row0 + lo) * K + 2 * hi;
  const float* wr = W + (size_t)(col0 + lo) * K + 2 * hi;

  v8f acc = {0.f, 0.f, 0.f, 0.f, 0.f, 0.f, 0.f, 0.f};
  #pragma unroll 8
  for (int kk = 0; kk < K; kk += 4) {
    v2f a = *(const v2f*)(xr + kk);
    v2f b = *(const v2f*)(wr + kk);
    acc = wmma_f32(a, b, acc);
  }

  const float bv = bias ? bias[col0 + lo] : 0.f;
  #pragma unroll
  for (int r = 0; r < 8; ++r)
    Y[(size_t)(row0 + r + 8 * hi) * Cout + col0 + lo] = acc[r] + bv;
}

// ---------------------------------------------------------------------------
// In-place per-head L2 norm along d=64; optionally folds exp(min(ls, MAX)).
// One thread per (b, h, row).
// ---------------------------------------------------------------------------
__global__ __launch_bounds__(256) void l2norm_head(
    float* __restrict__ buf, const float* __restrict__ logit_scale,
    int nrows, int rowStride, int useScale)
{
  const int t   = blockIdx.x * 256 + threadIdx.x;
  const int b   = t / (HH * nrows);
  const int rem = t - b * HH * nrows;
  const int h   = rem / nrows;
  const int n   = rem - h * nrows;
  float* p = buf + ((size_t)b * nrows + n) * rowStride + h * HD;

  float4 v[16];
  float ss = 0.f;
  #pragma unroll
  for (int i = 0; i < 16; ++i) {
    float4 x = *(const float4*)(p + 4 * i);
    v[i] = x;
    ss += x.x * x.x + x.y * x.y + x.z * x.z + x.w * x.w;
  }
  const float sc  = useScale ? __expf(fminf(logit_scale[h], LOGIT_MAX_C)) : 1.0f;
  const float inv = sc / fmaxf(sqrtf(ss), 1e-12f);
  #pragma unroll
  for (int i = 0; i < 16; ++i) {
    float4 x = v[i];
    x.x *= inv; x.y *= inv; x.z *= inv; x.w *= inv;
    *(float4*)(p + 4 * i) = x;
  }
}

// ---------------------------------------------------------------------------
// Fused flash attention with alibi + padding mask.
// Grid: (B*H, N/64). 128 threads = 4 waves; wave w owns q rows
// [qt*64 + w*16, +16). M streamed in 16-key tiles; K/V staged in LDS with
// GLOBAL_LOAD_ASYNC_TO_LDS_B128 (ASYNCcnt).
// ---------------------------------------------------------------------------
#define KV_STRIDE 68   // 68 floats = 272 B row stride (16B-aligned for B128)

__global__ __launch_bounds__(128) void attn_fused(
    const float* __restrict__ qn,    // [B,N,C]  (normalized, scale folded)
    const float* __restrict__ kvp,   // [B,M,2C] (k normalized, v raw)
    const float* __restrict__ alibi, // [B,H,N,M]
    const unsigned char* __restrict__ pmask, // [B,M] bool
    float* __restrict__ xo)          // [B,N,C]
{
  const int bh = blockIdx.x;
  const int b  = bh >> 4;          // / HH
  const int h  = bh & 15;
  const int qt = blockIdx.y;
  const int tid  = threadIdx.x;
  const int wave = tid >> 5;
  const int lane = tid & 31;
  const int lo   = lane & 15;
  const int hi   = lane >> 4;
  const int qrow0 = qt * 64 + wave * 16;

  __shared__ float Kl[16 * KV_STRIDE];   // 16 keys x 64 feats (padded)
  __shared__ float Vl[16 * KV_STRIDE];
  __shared__ float Pl[4][16 * 18];       // per-wave P transpose scratch

  // Preload this wave's 16 q rows in WMMA A-layout: qreg[t] = Q[lo][4t+2hi .. +1]
  v2f qreg[16];
  {
    const float* qb = qn + ((size_t)b * NN + qrow0 + lo) * CC + h * HD + 2 * hi;
    #pragma unroll
    for (int t = 0; t < 16; ++t) qreg[t] = *(const v2f*)(qb + 4 * t);
  }

  float mrow[8], lrow[8];
  v8f o[4];
  #pragma unroll
  for (int j = 0; j < 4; ++j) o[j] = (v8f){0.f,0.f,0.f,0.f,0.f,0.f,0.f,0.f};
  #pragma unroll
  for (int r = 0; r < 8; ++r) { mrow[r] = -1e30f; lrow[r] = 0.f; }

  const float* kb = kvp + (size_t)b * MM * (2 * CC) + h * HD;
  const float* vb = kb + CC;
  const float* ab = alibi + ((size_t)bh * NN + qrow0) * MM;
  const unsigned char* mb = pmask + (size_t)b * MM;

  for (int m0 = 0; m0 < MM; m0 += 16) {
    // ---- async K/V staging: 128 threads x 4 B128 transfers ----
    {
      const int r = tid >> 3;
      const int c = (tid & 7) * 8;
      const float* krow = kb + (size_t)(m0 + r) * (2 * CC) + c;
      const float* vrow = vb + (size_t)(m0 + r) * (2 * CC) + c;
      async_copy_b128(&Kl[r * KV_STRIDE + c],     krow);
      async_copy_b128(&Kl[r * KV_STRIDE + c + 4], krow + 4);
      async_copy_b128(&Vl[r * KV_STRIDE + c],     vrow);
      async_copy_b128(&Vl[r * KV_STRIDE + c + 4], vrow + 4);
      wait_async0();
    }
    __syncthreads();

    // ---- S = Q * K^T  (16x16 f32, 16 WMMAs over d=64) ----
    v8f s = (v8f){0.f,0.f,0.f,0.f,0.f,0.f,0.f,0.f};
    #pragma unroll
    for (int t = 0; t < 16; ++t) {
      v2f bk = *(const v2f*)&Kl[lo * KV_STRIDE + 4 * t + 2 * hi];
      s = wmma_f32(qreg[t], bk, s);
    }

    // ---- alibi + mask + online softmax (rows r+8*hi, column lo) ----
    const bool masked = (mb[m0 + lo] != 0);
    #pragma unroll
    for (int r = 0; r < 8; ++r) {
      const float raw = s[r] + ab[(size_t)(r + 8 * hi) * MM + m0 + lo];
      const float sv  = masked ? -3.0e38f : raw;

      float t = sv;                       // row max across 16 lanes
      t = fmaxf(t, __shfl_xor(t, 1, 32));
      t = fmaxf(t, __shfl_xor(t, 2, 32));
      t = fmaxf(t, __shfl_xor(t, 4, 32));
      t = fmaxf(t, __shfl_xor(t, 8, 32));
      const float mnew  = fmaxf(mrow[r], t);
      const float alpha = __expf(mrow[r] - mnew);
      mrow[r] = mnew;

      const float pv = __expf(sv - mnew);
      float ps = pv;                      // row sum across 16 lanes
      ps += __shfl_xor(ps, 1, 32);
      ps += __shfl_xor(ps, 2, 32);
      ps += __shfl_xor(ps, 4, 32);
      ps += __shfl_xor(ps, 8, 32);
      lrow[r] = lrow[r] * alpha + ps;

      #pragma unroll
      for (int j = 0; j < 4; ++j) o[j][r] *= alpha;
      Pl[wave][(r + 8 * hi) * 18 + lo] = pv;   // C/D -> [m][k] in LDS
    }
    // Same-wave LDS RAW: wait for our own ds stores before A-layout reads.
    asm volatile("s_wait_dscnt 0" ::: "memory");

    // ---- O += P @ V : 4 feature sub-tiles x 4 K-chunks = 16 WMMAs ----
    #pragma unroll
    for (int kk = 0; kk < 16; kk += 4) {
      v2f a = *(const v2f*)&Pl[wave][lo * 18 + kk + 2 * hi];
      #pragma unroll
      for (int j = 0; j < 4; ++j) {
        v2f bv;
        bv.x = Vl[(kk + 2 * hi)     * KV_STRIDE + j * 16 + lo];
        bv.y = Vl[(kk + 2 * hi + 1) * KV_STRIDE + j * 16 + lo];
        o[j] = wmma_f32(a, bv, o[j]);
      }
    }

    if (m0 + 16 < MM) {  // uniform branch: EXEC stays all-ones for WMMA
      __builtin_prefetch(&ab[(size_t)(8 * hi) * MM + m0 + 16 + lo], 0, 0);
    }
    __syncthreads();
  }

  // ---- epilogue: O / l -> x[b, n, h*64 + feat] ----
  float* ob = xo + ((size_t)b * NN + qrow0) * CC + h * HD;
  #pragma unroll
  for (int r = 0; r < 8; ++r) {
    const float inv = 1.0f / fmaxf(lrow[r], 1e-30f);
    #pragma unroll
    for (int j = 0; j < 4; ++j)
      ob[(size_t)(r + 8 * hi) * CC + j * 16 + lo] = o[j][r] * inv;
  }
}

// ---------------------------------------------------------------------------
// Launch: qproj -> kvproj -> norms -> fused attention -> out proj (+bias)
// Workspace: qp [B*N*C] | kvp [B*M*2C] | xat [B*N*C]  (48 MB f32)
// ---------------------------------------------------------------------------
extern "C" void kernel_launch(void* const* d_in, const int* in_sizes, int n_in,
                              void* d_out, int out_size, void* d_ws, size_t ws_size,
                              hipStream_t stream) {
  const float* q     = (const float*)d_in[0];   // [B,N,C]
  const float* kv    = (const float*)d_in[1];   // [B,M,C]
  const unsigned char* pad = (const unsigned char*)d_in[2]; // [B,M] bool (1B)
  const float* alibi = (const float*)d_in[3];   // [B,H,N,M]
  const float* Wq    = (const float*)d_in[4];   // [C,C]
  const float* Wkv   = (const float*)d_in[5];   // [2C,C]
  const float* Wo    = (const float*)d_in[6];   // [C,C]
  const float* bo    = (const float*)d_in[7];   // [C]
  const float* ls    = (const float*)d_in[8];   // [H,1,1]
  float* out = (float*)d_out;

  float* qp  = (float*)d_ws;                          // B*N*C
  float* kvp = qp  + (size_t)BB * NN * CC;            // B*M*2C
  float* xat = kvp + (size_t)BB * MM * 2 * CC;        // B*N*C

  const dim3 blk(128);

  // q projection: [B*N, C] x Wq[C,C]^T
  gemm16_nt<<<dim3(CC / 64, (BB * NN) / 16), blk, 0, stream>>>(
      q, Wq, nullptr, qp, CC, CC);
  // kv projection: [B*M, C] x Wkv[2C,C]^T
  gemm16_nt<<<dim3((2 * CC) / 64, (BB * MM) / 16), blk, 0, stream>>>(
      kv, Wkv, nullptr, kvp, CC, 2 * CC);

  // normalize q (fold logit scale) and k (first C of kvp rows)
  l2norm_head<<<(BB * HH * NN) / 256, 256, 0, stream>>>(qp, ls, NN, CC, 1);
  l2norm_head<<<(BB * HH * MM) / 256, 256, 0, stream>>>(kvp, ls, MM, 2 * CC, 0);

  // fused attention
  attn_fused<<<dim3(BB * HH, NN / 64), blk, 0, stream>>>(
      qp, kvp, alibi, pad, xat);

  // output projection with bias
  gemm16_nt<<<dim3(CC / 64, (BB * NN) / 16), blk, 0, stream>>>(
      xat, Wo, bo, out, CC, CC);
}